// ComboSumModule_25314537242674
// MI455X (gfx1250) — compile-verified
//
#include <hip/hip_runtime.h>

// CDNA5 / gfx1250 wave32
typedef __attribute__((ext_vector_type(2))) float v2f;
typedef __attribute__((ext_vector_type(8))) float v8f;

// Column-sum of one (batch, 16-column) strip using the matrix pipe:
// D = ones(16x4) * B(4x16) + C  =>  D[m,n] = sum_k B[k,n] + C[m,n].
// B layout (32-bit B 4x16): v0 = row k+0 (lanes 0-15) / k+2 (lanes 16-31),
//                           v1 = row k+1 / k+3, column = n0 + (lane & 15).
// After the K loop, C vgpr0 lanes 0-15 hold the 16 column sums (row M=0).
template <int K, int N>
__device__ __forceinline__ void wmma_strip_sum(const float* __restrict__ x,
                                               float* __restrict__ out,
                                               int batch, int n0, int lane) {
    const float* p = x + (size_t)batch * (K * N) + n0 + (lane & 15)
                       + ((lane < 16) ? 0 : 2) * N;   // fold rbase into base ptr
    v8f c = {0.f, 0.f, 0.f, 0.f, 0.f, 0.f, 0.f, 0.f};
    const v2f a = {1.0f, 1.0f};                        // all-ones A (16x4)
#pragma unroll 4
    for (int k = 0; k < K; k += 4) {
        v2f b;
        // streaming data, read once: non-temporal (TH=NT) loads
        b.x = __builtin_nontemporal_load(p + (size_t)k * N);
        b.y = __builtin_nontemporal_load(p + (size_t)(k + 1) * N);
        c = __builtin_amdgcn_wmma_f32_16x16x4_f32(
                /*neg_a=*/false, a, /*neg_b=*/false, b,
                /*c_mod=*/(short)0, c, /*reuse_a=*/false, /*reuse_b=*/false);
    }
    if (lane < 16)
        out[(size_t)batch * N + n0 + lane] = c[0];     // row M=0 = column sums
}

// Strip map (one wave per strip; all waves full, EXEC all-ones for WMMA):
//  x0: K=128 N=64 -> 4 strips/batch : waves [    0,  8192)
//  x1: K= 32 N=32 -> 2 strips/batch : waves [ 8192, 12288)
//  x2: K= 64 N=64 -> 4 strips/batch : waves [12288, 20480)
//  x3: K= 16 N=48 -> 3 strips/batch : waves [20480, 26624)
//  x4: K=200 N=32 -> 2 strips/batch : waves [26624, 30720)
//  x5: K=  8 N= 8 -> scalar tail    : waves [30720, 31232)
// Output offsets (floats): 0, 131072, 196608, 327680, 425984, 491520 (total 507904)
__global__ __launch_bounds__(256) void combo_sum_kernel(
    const float* __restrict__ x0, const float* __restrict__ x1,
    const float* __restrict__ x2, const float* __restrict__ x3,
    const float* __restrict__ x4, const float* __restrict__ x5,
    float* __restrict__ out) {
    const int tid  = blockIdx.x * blockDim.x + threadIdx.x;
    const int wave = tid >> 5;   // uniform within a wave
    const int lane = tid & 31;

    if (wave < 8192) {
        const int s = wave;
        wmma_strip_sum<128, 64>(x0, out + 0,      s >> 2, (s & 3) << 4, lane);
    } else if (wave < 12288) {
        const int s = wave - 8192;
        wmma_strip_sum<32, 32>(x1, out + 131072,  s >> 1, (s & 1) << 4, lane);
    } else if (wave < 20480) {
        const int s = wave - 12288;
        wmma_strip_sum<64, 64>(x2, out + 196608,  s >> 2, (s & 3) << 4, lane);
    } else if (wave < 26624) {
        const int s = wave - 20480;
        wmma_strip_sum<16, 48>(x3, out + 327680,  s / 3, (s % 3) << 4, lane);
    } else if (wave < 30720) {
        const int s = wave - 26624;
        wmma_strip_sum<200, 32>(x4, out + 425984, s >> 1, (s & 1) << 4, lane);
    } else {
        // x5: 2048 x 8 x 8, N=8 not WMMA-shaped; tiny (0.5 MB) scalar path.
        const int idx = ((wave - 30720) << 5) + lane;  // [0, 16384)
        const int b = idx >> 3, col = idx & 7;
        const float* p = x5 + b * 64 + col;
        float s = 0.f;
#pragma unroll
        for (int k = 0; k < 8; ++k)
            s += __builtin_nontemporal_load(p + k * 8);
        out[491520 + idx] = s;
    }
}

extern "C" void kernel_launch(void* const* d_in, const int* in_sizes, int n_in,
                              void* d_out, int out_size, void* d_ws, size_t ws_size,
                              hipStream_t stream) {
    const float* x0 = (const float*)d_in[0];
    const float* x1 = (const float*)d_in[1];
    const float* x2 = (const float*)d_in[2];
    const float* x3 = (const float*)d_in[3];
    const float* x4 = (const float*)d_in[4];
    const float* x5 = (const float*)d_in[5];
    // d_in[6] is `dim` == 1, baked into the kernel.
    float* out = (float*)d_out;

    // 31232 waves total = 3904 blocks x 256 threads (8 waves/block).
    combo_sum_kernel<<<3904, 256, 0, stream>>>(x0, x1, x2, x3, x4, x5, out);
}